// DotProductAttentionStream_14319420964899
// MI455X (gfx1250) — compile-verified
//
#include <hip/hip_runtime.h>

typedef __attribute__((ext_vector_type(2))) float v2f;
typedef __attribute__((ext_vector_type(8))) float v8f;

#define ALPHA_ 0.99f

constexpr int B_  = 64;
constexpr int N1_ = 64;
constexpr int N2_ = 512;
constexpr int D_  = 64;
constexpr int T_  = 128;
constexpr int QKS = N2_ + 4;   // 516: qk tile row stride (floats)
constexpr int QS  = D_ + 4;    // 68:  q tile row stride (floats)
constexpr size_t SBD = (size_t)B_ * N1_ * D_;  // stride between t-slices of out

__global__ __launch_bounds__(128) void attn_stream_kernel(
    const float* __restrict__ q, const float* __restrict__ k_init,
    const float* __restrict__ v_init, const float* __restrict__ mask,
    const float* __restrict__ ks, const float* __restrict__ vs,
    float* __restrict__ out)
{
  __shared__ __align__(16) float q_lds[16 * QS];     // Q tile 16x64
  __shared__ __align__(16) float qk_lds[16 * QKS];   // QK / QK_exp tile 16x512
  __shared__ float pmax[16 * 8];
  __shared__ float psum[16 * 8];
  __shared__ float rowmax[16];
  __shared__ float rowsum[16];
  __shared__ __align__(16) float kt_lds[64];
  __shared__ __align__(16) float vt_lds[64];

  const int tid   = threadIdx.x;
  const int wave  = tid >> 5;
  const int lane  = tid & 31;
  const int lrow  = lane & 15;   // row / N index within 16
  const int lhalf = lane >> 4;   // lane half selects K pair / M+8
  const int b     = blockIdx.x >> 2;
  const int row0  = (blockIdx.x & 3) * 16;

  // ---- load Q tile (16 x 64) into LDS ----
  {
    const float4* qg4 = (const float4*)(q + ((size_t)b * N1_ + row0) * D_);
    for (int v = tid; v < 256; v += 128) {
      int r = v >> 4, c = v & 15;
      float4 val = qg4[v];
      *(float4*)&q_lds[r * QS + c * 4] = val;
    }
  }
  __syncthreads();

  // ---- phase 1a: QK[16 x 512] = Q @ K^T ; wave w covers cols [128w,128w+128) ----
  for (int j = 0; j < 8; ++j) {
    const int m0 = wave * 128 + j * 16;
    const float* kb = k_init + ((size_t)b * N2_ + m0 + lrow) * D_ + 2 * lhalf;
    if (j < 7) __builtin_prefetch(kb + 16 * D_, 0, 0);   // gfx1250 global_prefetch_b8
    v8f acc = {0.f, 0.f, 0.f, 0.f, 0.f, 0.f, 0.f, 0.f};
    #pragma unroll
    for (int k0 = 0; k0 < D_; k0 += 4) {
      // A: 16x4 fp32, lane holds Q[row=lrow][k0+2*lhalf .. +1]
      v2f a = *(const v2f*)&q_lds[lrow * QS + k0 + 2 * lhalf];
      // B: 4x16 fp32, B[K][N] = K_init[m0+N][k0+K]; lane holds contiguous d-pair
      v2f bb = *(const v2f*)(kb + k0);
      acc = __builtin_amdgcn_wmma_f32_16x16x4_f32(
          false, a, false, bb, (short)0, acc, false, false);
    }
    #pragma unroll
    for (int r = 0; r < 8; ++r)
      qk_lds[(r + 8 * lhalf) * QKS + m0 + lrow] = acc[r];
  }
  __syncthreads();

  // ---- phase 1b: row max over N2 (16 rows x 8 partial reducers) ----
  {
    const int rr = tid & 15, sub = tid >> 4;
    const float* rp = &qk_lds[rr * QKS + sub * 64];
    float m = -3.4e38f;
    for (int c = 0; c < 64; ++c) m = fmaxf(m, rp[c]);
    pmax[rr * 8 + sub] = m;
  }
  __syncthreads();
  if (tid < 16) {
    float m = pmax[tid * 8];
    for (int s = 1; s < 8; ++s) m = fmaxf(m, pmax[tid * 8 + s]);
    rowmax[tid] = m;
  }
  __syncthreads();

  // ---- phase 1c: QK_exp = exp(QK - max + mask), and row sums Z ----
  {
    const int rr = tid & 15, sub = tid >> 4;
    float* rp = &qk_lds[rr * QKS + sub * 64];
    const float* mp = mask + ((size_t)b * N1_ + row0 + rr) * N2_ + sub * 64;
    const float rmax = rowmax[rr];
    float s = 0.f;
    for (int c = 0; c < 64; ++c) {
      float e = __expf(rp[c] - rmax + mp[c]);
      rp[c] = e;
      s += e;
    }
    psum[rr * 8 + sub] = s;
  }
  __syncthreads();
  if (tid < 16) {
    float s = 0.f;
    for (int ss = 0; ss < 8; ++ss) s += psum[tid * 8 + ss];
    rowsum[tid] = s;
  }
  __syncthreads();

  // ---- phase 1d: QKV[16 x 64] = QK_exp @ V ; wave w covers D cols [16w,16w+16) ----
  const int n0 = wave * 16;
  v8f acc = {0.f, 0.f, 0.f, 0.f, 0.f, 0.f, 0.f, 0.f};   // QKV carry (C layout)
  {
    const float* vb = v_init + (size_t)b * N2_ * D_ + n0 + lrow;
    for (int k0 = 0; k0 < N2_; k0 += 4) {
      // A: QK_exp[row=lrow][k0+2*lhalf .. +1]
      v2f a = *(const v2f*)&qk_lds[lrow * QKS + k0 + 2 * lhalf];
      // B: B[K][N] = V[m=k0+K][n0+N] (stride-D_ pair of loads)
      v2f bv;
      bv.x = vb[(k0 + 2 * lhalf) * D_];
      bv.y = vb[(k0 + 2 * lhalf + 1) * D_];
      acc = __builtin_amdgcn_wmma_f32_16x16x4_f32(
          false, a, false, bv, (short)0, acc, false, false);
    }
  }

  // ---- out0 = QKV / Z ----
  {
    float* op = out + ((size_t)b * N1_ + row0) * D_ + n0 + lrow;
    #pragma unroll
    for (int r = 0; r < 8; ++r) {
      const int M = r + 8 * lhalf;
      op[(size_t)M * D_] = __fdividef(acc[r], rowsum[M]);
    }
  }
  float zl = rowsum[lrow];          // per-lane Z carry for row lrow
  const float rmax_l = rowmax[lrow];
  __syncthreads();

  // ---- phase 2: streaming rank-1 pushes with alpha decay ----
  const float* ksb = ks + (size_t)b * D_;
  const float* vsb = vs + (size_t)b * D_;
  for (int t = 0; t < T_; ++t) {
    if (tid < 64) kt_lds[tid]      = ksb[(size_t)t * (B_ * D_) + tid];
    else          vt_lds[tid - 64] = vsb[(size_t)t * (B_ * D_) + (tid - 64)];
    __syncthreads();

    // qk[row] = q[row,:] . k_t  (lanes m and m+16 redundantly compute row m)
    float qk = 0.f;
    #pragma unroll
    for (int d4 = 0; d4 < 16; ++d4) {
      float4 qv = *(const float4*)&q_lds[lrow * QS + d4 * 4];
      float4 kv = *(const float4*)&kt_lds[d4 * 4];
      qk = fmaf(qv.x, kv.x, qk); qk = fmaf(qv.y, kv.y, qk);
      qk = fmaf(qv.z, kv.z, qk); qk = fmaf(qv.w, kv.w, qk);
    }
    float e = __expf(qk - rmax_l);
    zl = ALPHA_ * zl + e;
    const float vval = vt_lds[n0 + lrow];

    float* op = out + (size_t)(t + 1) * SBD + ((size_t)b * N1_ + row0) * D_ + n0 + lrow;
    #pragma unroll
    for (int r = 0; r < 8; ++r) {
      const int src = r + 8 * lhalf;            // lane holding row M = r + 8*lhalf
      float er = __shfl(e,  src, 32);
      float zr = __shfl(zl, src, 32);
      acc[r] = ALPHA_ * acc[r] + er * vval;
      op[(size_t)(r + 8 * lhalf) * D_] = __fdividef(acc[r], zr);
    }
    __syncthreads();
  }
}

extern "C" void kernel_launch(void* const* d_in, const int* in_sizes, int n_in,
                              void* d_out, int out_size, void* d_ws, size_t ws_size,
                              hipStream_t stream) {
  (void)in_sizes; (void)n_in; (void)out_size; (void)d_ws; (void)ws_size;
  const float* q      = (const float*)d_in[0];
  const float* k_init = (const float*)d_in[1];
  const float* v_init = (const float*)d_in[2];
  const float* mask   = (const float*)d_in[3];
  const float* k_strm = (const float*)d_in[4];
  const float* v_strm = (const float*)d_in[5];
  float* out = (float*)d_out;

  dim3 grid(B_ * (N1_ / 16));   // 256 workgroups: (batch, n1-tile)
  dim3 block(128);              // 4 wave32 waves
  attn_stream_kernel<<<grid, block, 0, stream>>>(q, k_init, v_init, mask,
                                                 k_strm, v_strm, out);
}